// DSVDD_61392262529254
// MI455X (gfx1250) — compile-verified
//
#include <hip/hip_runtime.h>
#include <math.h>

// ---------------------------------------------------------------------------
// DSVDD scoring pipeline for MI455X (gfx1250, wave32, WMMA).
// Both GEMMs (~110 GFLOP fp32) run on V_WMMA_F32_16X16X4_F32. The distance
// GEMM is fused with the top-3 / softmin reduction (never materializes the
// 315MB distance tensor). B matrices are pre-packed K-pair-interleaved so
// every B fragment is one aligned global_load_b64; the distance kernel
// groups 4 N-tiles per pass so each A fragment feeds 4 WMMAs.
// ---------------------------------------------------------------------------

typedef __attribute__((ext_vector_type(2))) float v2f;
typedef __attribute__((ext_vector_type(8))) float v8f;

#define BB    8
#define HH    56
#define WWD   56
#define HWN   (HH * WWD)      // 3136
#define DIN   1794
#define DINP  1796            // padded to multiple of 4 (WMMA K-step)
#define KP    (DINP / 2)      // 898 K-pairs (conv B)
#define DOUT  448
#define KP2   (DOUT / 2)      // 224 K-pairs (dist B)
#define NCEN  3136
#define MTOT  (BB * HWN)      // 25088

// fp32 WMMA: D(16x16) = A(16x4) * B(4x16) + C.
// A frag: lane l holds A[l&15][2*(l>>4) + {0,1}]  (v2f)
// B frag: lane l holds B[2*(l>>4) + {0,1}][l&15]  (v2f from packed pairs)
// C/D:    lane l, slot j holds D[j + 8*(l>>4)][l&15]
__device__ __forceinline__ v8f wmma4(v2f a, v2f b, v8f c) {
    return __builtin_amdgcn_wmma_f32_16x16x4_f32(
        /*neg_a=*/false, a, /*neg_b=*/false, b,
        /*c_mod=*/(short)0, c, /*reuse_a=*/false, /*reuse_b=*/false);
}

// ---------------------------------------------------------------------------
// Kernel 1: 3x3 avg pool, stride 1, zero pad, count_include_pad (/9 always)
// ---------------------------------------------------------------------------
__global__ void pool3_kernel(const float* __restrict__ in, float* __restrict__ out,
                             int C, int Hp, int Wp) {
    long idx = (long)blockIdx.x * blockDim.x + threadIdx.x;
    long total = (long)BB * C * Hp * Wp;
    if (idx >= total) return;
    int x  = (int)(idx % Wp);
    int y  = (int)((idx / Wp) % Hp);
    long bc = idx / ((long)Wp * Hp);
    const float* p = in + bc * (long)Hp * Wp;
    float s = 0.0f;
    #pragma unroll
    for (int dy = -1; dy <= 1; ++dy) {
        int yy = y + dy;
        if ((unsigned)yy >= (unsigned)Hp) continue;
        #pragma unroll
        for (int dx = -1; dx <= 1; ++dx) {
            int xx = x + dx;
            if ((unsigned)xx >= (unsigned)Wp) continue;
            s += p[yy * Wp + xx];
        }
    }
    out[idx] = s * (1.0f / 9.0f);
}

// ---------------------------------------------------------------------------
// Kernel 2: build padded sample matrix [M=25088][K=1796] row-major:
//   c<256 : pooled L0 | 256-767 : bilerp pooled L1 | 768-1791 : bilerp L2
//   1792 : xx coord (row) | 1793 : yy coord (col) | 1794-95 : zero pad
// half-pixel-center bilinear with edge clamp: src = (dst+0.5)*scale - 0.5
// ---------------------------------------------------------------------------
__device__ __forceinline__ float bilerp(const float* __restrict__ m, int S,
                                        int h, int w, float sc, float off) {
    float fy = (float)h * sc + off;
    float fx = (float)w * sc + off;
    int y0 = (int)floorf(fy), x0 = (int)floorf(fx);
    float wy = fy - (float)y0, wx = fx - (float)x0;
    int y0c = y0 < 0 ? 0 : y0;
    int y1c = (y0 + 1) > (S - 1) ? (S - 1) : (y0 + 1);
    int x0c = x0 < 0 ? 0 : x0;
    int x1c = (x0 + 1) > (S - 1) ? (S - 1) : (x0 + 1);
    float v00 = m[y0c * S + x0c], v01 = m[y0c * S + x1c];
    float v10 = m[y1c * S + x0c], v11 = m[y1c * S + x1c];
    return (1.0f - wy) * ((1.0f - wx) * v00 + wx * v01) +
           wy * ((1.0f - wx) * v10 + wx * v11);
}

__global__ void build_sample_kernel(const float* __restrict__ pool0,
                                    const float* __restrict__ pool1,
                                    const float* __restrict__ pool2,
                                    float* __restrict__ sample) {
    long idx = (long)blockIdx.x * blockDim.x + threadIdx.x;
    if (idx >= (long)MTOT * DINP) return;
    int c   = (int)(idx % DINP);
    long mh = idx / DINP;              // b*HW + hw
    int hw  = (int)(mh % HWN);
    int b   = (int)(mh / HWN);
    int h = hw / WWD, w = hw % WWD;
    float v;
    if (c < 256) {
        v = pool0[(((size_t)b * 256 + c) * HH + h) * WWD + w];
    } else if (c < 768) {
        v = bilerp(pool1 + ((size_t)b * 512 + (c - 256)) * 28 * 28, 28, h, w,
                   0.5f, -0.25f);
    } else if (c < 1792) {
        v = bilerp(pool2 + ((size_t)b * 1024 + (c - 768)) * 14 * 14, 14, h, w,
                   0.25f, -0.375f);
    } else if (c == 1792) {
        v = (float)h * (2.0f / 55.0f) - 1.0f;   // xx: varies along H
    } else if (c == 1793) {
        v = (float)w * (2.0f / 55.0f) - 1.0f;   // yy: varies along W
    } else {
        v = 0.0f;                                // K padding
    }
    sample[idx] = v;
}

// ---------------------------------------------------------------------------
// Kernel 3: conv_w [448][1794] -> Wtp [898 pairs][448] v2f, K-pair packed:
// Wtp[p][n] = { W^T[2p][n], W^T[2p+1][n] }  (zero pad K >= 1794)
// ---------------------------------------------------------------------------
__global__ void pack_wt_kernel(const float* __restrict__ cw, v2f* __restrict__ wtp) {
    int idx = blockIdx.x * blockDim.x + threadIdx.x;
    if (idx >= KP * DOUT) return;
    int n = idx % DOUT;
    int p = idx / DOUT;
    int k0 = 2 * p, k1 = 2 * p + 1;
    v2f v;
    v.x = (k0 < DIN) ? cw[(size_t)n * DIN + k0] : 0.0f;
    v.y = (k1 < DIN) ? cw[(size_t)n * DIN + k1] : 0.0f;
    wtp[idx] = v;
}

// ---------------------------------------------------------------------------
// Kernel 4: C [448][3136] -> Cp [224 pairs][3136] v2f, K-pair packed
// ---------------------------------------------------------------------------
__global__ void pack_c_kernel(const float* __restrict__ Cm, v2f* __restrict__ cp) {
    int idx = blockIdx.x * blockDim.x + threadIdx.x;
    if (idx >= KP2 * NCEN) return;
    int n = idx % NCEN;
    int p = idx / NCEN;
    v2f v;
    v.x = Cm[(size_t)(2 * p) * NCEN + n];
    v.y = Cm[(size_t)(2 * p + 1) * NCEN + n];
    cp[idx] = v;
}

// ---------------------------------------------------------------------------
// Kernel 5: cent2[n] = sum_k C[k][n]^2
// ---------------------------------------------------------------------------
__global__ void cent2_kernel(const float* __restrict__ Cm, float* __restrict__ c2) {
    int n = blockIdx.x * blockDim.x + threadIdx.x;
    if (n >= NCEN) return;
    float s = 0.0f;
    for (int k = 0; k < DOUT; ++k) {
        float v = Cm[(size_t)k * NCEN + n];
        s += v * v;
    }
    c2[n] = s;
}

// ---------------------------------------------------------------------------
// Kernel 6: conv GEMM  phi[25088][448] = sample[25088][1796] * W^T + bias
// Block = 8 waves; wave owns 16 rows; block covers 64 cols (4 accumulators).
// Per k-step: 1 A b64 + 4 B b64 feed 4 WMMAs (1.25 loads/WMMA).
// ---------------------------------------------------------------------------
__global__ __launch_bounds__(256) void conv_gemm_kernel(
        const float* __restrict__ A, const v2f* __restrict__ Bp,
        const float* __restrict__ bias, float* __restrict__ phi) {
    const int lane  = threadIdx.x & 31;
    const int wave  = threadIdx.x >> 5;
    const int khalf = lane >> 4;           // 0/1 -> K-pair select
    const int l15   = lane & 15;
    const int rowBase = blockIdx.x * 128 + wave * 16;
    const int colBase = blockIdx.y * 64;

    v8f acc0 = {}, acc1 = {}, acc2 = {}, acc3 = {};
    const float* arow = A + (size_t)(rowBase + l15) * DINP;

    for (int k0 = 0; k0 < DINP; k0 += 4) {
        v2f a = *(const v2f*)(arow + k0 + 2 * khalf);     // 8B aligned
        const v2f* bp = Bp + (size_t)((k0 >> 1) + khalf) * DOUT + colBase + l15;
        acc0 = wmma4(a, bp[0],  acc0);
        acc1 = wmma4(a, bp[16], acc1);
        acc2 = wmma4(a, bp[32], acc2);
        acc3 = wmma4(a, bp[48], acc3);
    }

    const float bv0 = bias[colBase + l15];
    const float bv1 = bias[colBase + l15 + 16];
    const float bv2 = bias[colBase + l15 + 32];
    const float bv3 = bias[colBase + l15 + 48];
    #pragma unroll
    for (int j = 0; j < 8; ++j) {
        int r = rowBase + j + 8 * khalf;
        float* prow = phi + (size_t)r * DOUT + colBase + l15;
        prow[0]  = acc0[j] + bv0;
        prow[16] = acc1[j] + bv1;
        prow[32] = acc2[j] + bv2;
        prow[48] = acc3[j] + bv3;
    }
}

// ---------------------------------------------------------------------------
// Kernel 7: feat2[m] = sum_o phi[m][o]^2
// ---------------------------------------------------------------------------
__global__ void feat2_kernel(const float* __restrict__ phi, float* __restrict__ f2) {
    int m = blockIdx.x * blockDim.x + threadIdx.x;
    if (m >= MTOT) return;
    const float* p = phi + (size_t)m * DOUT;
    float s = 0.0f;
    for (int o = 0; o < DOUT; ++o) s += p[o] * p[o];
    f2[m] = s;
}

// ---------------------------------------------------------------------------
// Kernel 8: fused distance GEMM + top-3 + softmin.
// Wave owns 16 phi-rows; streams 49 groups of 4 center tiles (64 cols) with
// 4 accumulators so each A fragment feeds 4 WMMAs. Per lane keeps a running
// top-3 over its 8 (row,colclass) slots; cross-lane merge via LDS.
// dist = sqrt(feat2 + cent2 - 2*phi.C); score = softmax(-top3)[0]*top3[0].
// ---------------------------------------------------------------------------
__device__ __forceinline__ void ins3(float d, float& t0, float& t1, float& t2) {
    if (d < t0)      { t2 = t1; t1 = t0; t0 = d; }
    else if (d < t1) { t2 = t1; t1 = d; }
    else if (d < t2) { t2 = d; }
}

__global__ __launch_bounds__(256) void dist_topk_kernel(
        const float* __restrict__ phi, const v2f* __restrict__ Cp,
        const float* __restrict__ f2, const float* __restrict__ c2,
        float* __restrict__ out) {
    __shared__ float red[8][16][16][3];   // [wave][row][col-lane][rank] = 24KB
    const int lane  = threadIdx.x & 31;
    const int wave  = threadIdx.x >> 5;
    const int khalf = lane >> 4;
    const int l15   = lane & 15;
    const int mBase = blockIdx.x * 128 + wave * 16;

    const float* arow = phi + (size_t)(mBase + l15) * DOUT;
    float f2r[8];
    #pragma unroll
    for (int j = 0; j < 8; ++j) f2r[j] = f2[mBase + j + 8 * khalf];

    float t0[8], t1[8], t2[8];
    #pragma unroll
    for (int j = 0; j < 8; ++j) { t0[j] = 3e38f; t1[j] = 3e38f; t2[j] = 3e38f; }

    for (int ng = 0; ng < NCEN / 64; ++ng) {          // 49 groups of 4 tiles
        const int n0 = ng * 64 + l15;
        if (ng + 1 < NCEN / 64)
            __builtin_prefetch(Cp + n0 + 64, 0, 0);    // global_prefetch_b8
        v8f acc0 = {}, acc1 = {}, acc2 = {}, acc3 = {};
        for (int k0 = 0; k0 < DOUT; k0 += 4) {
            v2f a = *(const v2f*)(arow + k0 + 2 * khalf);  // 8B aligned
            const v2f* bp = Cp + (size_t)((k0 >> 1) + khalf) * NCEN + n0;
            acc0 = wmma4(a, bp[0],  acc0);
            acc1 = wmma4(a, bp[16], acc1);
            acc2 = wmma4(a, bp[32], acc2);
            acc3 = wmma4(a, bp[48], acc3);
        }
        const float cn0 = c2[n0];
        const float cn1 = c2[n0 + 16];
        const float cn2 = c2[n0 + 32];
        const float cn3 = c2[n0 + 48];
        #pragma unroll
        for (int j = 0; j < 8; ++j) {
            float d;
            d = sqrtf(fmaxf(f2r[j] + cn0 - 2.0f * acc0[j], 0.0f)); ins3(d, t0[j], t1[j], t2[j]);
            d = sqrtf(fmaxf(f2r[j] + cn1 - 2.0f * acc1[j], 0.0f)); ins3(d, t0[j], t1[j], t2[j]);
            d = sqrtf(fmaxf(f2r[j] + cn2 - 2.0f * acc2[j], 0.0f)); ins3(d, t0[j], t1[j], t2[j]);
            d = sqrtf(fmaxf(f2r[j] + cn3 - 2.0f * acc3[j], 0.0f)); ins3(d, t0[j], t1[j], t2[j]);
        }
    }

    #pragma unroll
    for (int j = 0; j < 8; ++j) {
        int r = j + 8 * khalf;
        red[wave][r][l15][0] = t0[j];
        red[wave][r][l15][1] = t1[j];
        red[wave][r][l15][2] = t2[j];
    }
    __syncthreads();

    if (lane < 16) {
        float a0 = 3e38f, a1 = 3e38f, a2 = 3e38f;
        #pragma unroll 4
        for (int c = 0; c < 16; ++c) {
            ins3(red[wave][lane][c][0], a0, a1, a2);
            ins3(red[wave][lane][c][1], a0, a1, a2);
            ins3(red[wave][lane][c][2], a0, a1, a2);
        }
        // softmin over ascending (a0,a1,a2): s0 = 1/(1+e^(a0-a1)+e^(a0-a2))
        float e1 = __expf(a0 - a1), e2 = __expf(a0 - a2);
        out[mBase + lane] = a0 / (1.0f + e1 + e2);
    }
}

// ---------------------------------------------------------------------------
// Launch: all on `stream`, scratch carved out of d_ws (~234 MB of floats).
// phi aliases the pooled buffers (both exactly 11,239,424 floats) since the
// pools are dead once `sample` is built.
// ---------------------------------------------------------------------------
extern "C" void kernel_launch(void* const* d_in, const int* in_sizes, int n_in,
                              void* d_out, int out_size, void* d_ws, size_t ws_size,
                              hipStream_t stream) {
    (void)in_sizes; (void)n_in; (void)out_size; (void)ws_size;
    const float* p0     = (const float*)d_in[0];
    const float* p1     = (const float*)d_in[1];
    const float* p2     = (const float*)d_in[2];
    const float* conv_w = (const float*)d_in[3];
    const float* conv_b = (const float*)d_in[4];
    const float* Cm     = (const float*)d_in[5];
    float* out = (float*)d_out;

    float* ws = (float*)d_ws;
    const size_t nPool0 = (size_t)BB * 256 * HWN;        // 6,422,528
    const size_t nPool1 = (size_t)BB * 512 * 28 * 28;    // 3,211,264
    const size_t nPool2 = (size_t)BB * 1024 * 14 * 14;   // 1,605,632
    const size_t nPhi   = (size_t)MTOT * DOUT;           // 11,239,424 == pools
    const size_t nSamp  = (size_t)MTOT * DINP;           // 45,058,048
    const size_t nWtp   = (size_t)KP * DOUT * 2;         //    804,608 floats
    const size_t nCp    = (size_t)KP2 * NCEN * 2;        //  1,404,928 floats

    float* pool0  = ws;
    float* pool1  = pool0 + nPool0;
    float* pool2  = pool1 + nPool1;
    float* phi    = ws;                 // alias over pools (dead after sample)
    float* sample = ws + nPhi;
    v2f*   wtp    = (v2f*)(sample + nSamp);
    v2f*   cp     = (v2f*)(sample + nSamp + nWtp);
    float* feat2  = sample + nSamp + nWtp + nCp;
    float* cent2  = feat2 + MTOT;

    const int T = 256;
    // 1) avg-pool each pyramid level
    pool3_kernel<<<(int)((nPool0 + T - 1) / T), T, 0, stream>>>(p0, pool0, 256, HH, WWD);
    pool3_kernel<<<(int)((nPool1 + T - 1) / T), T, 0, stream>>>(p1, pool1, 512, 28, 28);
    pool3_kernel<<<(int)((nPool2 + T - 1) / T), T, 0, stream>>>(p2, pool2, 1024, 14, 14);
    // 2) descriptor: gather + bilinear + coords + zero-pad K
    build_sample_kernel<<<(int)((nSamp + T - 1) / T), T, 0, stream>>>(pool0, pool1, pool2, sample);
    // 3) pack W^T / C into K-pair-interleaved fragment layout, 4) center norms
    pack_wt_kernel<<<(KP * DOUT + T - 1) / T, T, 0, stream>>>(conv_w, wtp);
    pack_c_kernel<<<(KP2 * NCEN + T - 1) / T, T, 0, stream>>>(Cm, cp);
    cent2_kernel<<<(NCEN + T - 1) / T, T, 0, stream>>>(Cm, cent2);
    // 5) conv GEMM (WMMA fp32)
    conv_gemm_kernel<<<dim3(MTOT / 128, DOUT / 64), T, 0, stream>>>(sample, wtp, conv_b, phi);
    // 6) row norms
    feat2_kernel<<<(MTOT + T - 1) / T, T, 0, stream>>>(phi, feat2);
    // 7) fused distance GEMM + top-3 + softmin (WMMA fp32)
    dist_topk_kernel<<<MTOT / 128, T, 0, stream>>>(phi, cp, feat2, cent2, out);
}